// Codebook_3891240370351
// MI455X (gfx1250) — compile-verified
//
#include <hip/hip_runtime.h>

// Problem constants (B=32, D=64, H=32, W=32, K=1024)
#define D_     64
#define HW_    1024        // H*W
#define N_     32768       // B*H*W rows of zf
#define K_     1024
#define NELEM_ 2097152     // B*D*H*W

typedef __attribute__((ext_vector_type(2))) float v2f;
typedef __attribute__((ext_vector_type(8))) float v8f;

// ---------------------------------------------------------------------------
// Kernel 1: fused dot-product (f32 WMMA 16x16x4) + surrogate-distance argmin.
// One wave handles a 16-row strip of zf; loops over 64 tiles of 16 codes.
// dist surrogate = ||e||^2 - 2*dot (row norm dropped: constant per row).
// Two independent WMMA accumulator chains (even/odd k-steps) to break the
// serial WMMA->WMMA RAW dependency and double in-wave matrix-pipe ILP.
// ---------------------------------------------------------------------------
__global__ __launch_bounds__(256) void vq_argmin_kernel(
    const float* __restrict__ z, const float* __restrict__ emb,
    float* __restrict__ out_idx_f, int* __restrict__ ws_idx)
{
    const int tid   = blockIdx.x * blockDim.x + threadIdx.x;
    const int wave  = tid >> 5;
    const int lane  = tid & 31;
    const int laneM = lane & 15;   // row (A) / code col (B,C) within tile
    const int half  = lane >> 4;   // selects K pair {0,1} vs {2,3} per wmma step
    const int n0    = wave << 4;   // first of 16 rows for this wave
    const int n     = n0 + laneM;  // global zf row
    const int b     = n >> 10;     // n / HW
    const int hw    = n & (HW_ - 1);
    const float* zb = z + (size_t)b * (D_ * HW_) + hw;   // NCHW: + d*HW

    // A strip: 16x64 f32, per-lane 16 v2f per the f32 16x16x4 A layout:
    // lanes 0-15 hold K = 4s+{0,1}, lanes 16-31 hold K = 4s+{2,3}.
    v2f a[16];
#pragma unroll
    for (int s = 0; s < 16; ++s) {
        const int d0 = 4 * s + 2 * half;
        a[s].x = zb[(size_t)d0 * HW_];
        a[s].y = zb[(size_t)(d0 + 1) * HW_];
    }

    float bestd[8];
    int   besti[8];
#pragma unroll
    for (int v = 0; v < 8; ++v) { bestd[v] = 3.4e38f; besti[v] = 0; }

    for (int c0 = 0; c0 < K_; c0 += 16) {
        // B tile column = embedding row (c0 + laneM); per-lane half of its K's
        const float* erow = emb + (size_t)(c0 + laneM) * D_ + 2 * half;
        v8f   acc0 = {};
        v8f   acc1 = {};
        float psum = 0.0f;
#pragma unroll
        for (int s = 0; s < 16; s += 2) {
            v2f bv0, bv1;
            bv0.x = erow[4 * s];
            bv0.y = erow[4 * s + 1];
            bv1.x = erow[4 * s + 4];
            bv1.y = erow[4 * s + 5];
            psum += bv0.x * bv0.x + bv0.y * bv0.y;
            psum += bv1.x * bv1.x + bv1.y * bv1.y;
            acc0 = __builtin_amdgcn_wmma_f32_16x16x4_f32(
                false, a[s],     false, bv0, (short)0, acc0, false, false);
            acc1 = __builtin_amdgcn_wmma_f32_16x16x4_f32(
                false, a[s + 1], false, bv1, (short)0, acc1, false, false);
        }
        // lane L and L+16 hold complementary K halves of the same code column
        const float enorm = psum + __shfl_xor(psum, 16, 32);
        const int   code  = c0 + laneM;
#pragma unroll
        for (int v = 0; v < 8; ++v) {
            const float dot  = acc0[v] + acc1[v];       // row m = v + 8*half
            const float dsur = enorm - 2.0f * dot;
            if (dsur < bestd[v]) { bestd[v] = dsur; besti[v] = code; }
        }
    }

    // Reduce over the 16 code-columns per lane group (xor 1/2/4/8 stays in group)
#pragma unroll
    for (int off = 1; off < 16; off <<= 1) {
#pragma unroll
        for (int v = 0; v < 8; ++v) {
            const float od = __shfl_xor(bestd[v], off, 32);
            const int   oi = __shfl_xor(besti[v], off, 32);
            if (od < bestd[v] || (od == bestd[v] && oi < besti[v])) {
                bestd[v] = od; besti[v] = oi;
            }
        }
    }

    if (laneM == 0) {
        const int rbase = n0 + half * 8;   // lanes 0-15: rows 0-7, 16-31: rows 8-15
#pragma unroll
        for (int v = 0; v < 8; ++v) {
            ws_idx[rbase + v]    = besti[v];
            out_idx_f[rbase + v] = (float)besti[v];
        }
    }
}

// ---------------------------------------------------------------------------
// Kernel 2: gather + straight-through output (NCHW) + per-block sq-diff sums.
// ---------------------------------------------------------------------------
__global__ __launch_bounds__(256) void vq_quant_kernel(
    const float* __restrict__ z, const float* __restrict__ emb,
    const int* __restrict__ ws_idx, float* __restrict__ out,
    float* __restrict__ partial)
{
    __shared__ float sdata[256];
    const int e = blockIdx.x * 256 + threadIdx.x;   // NCHW flat index
    const int hw = e & (HW_ - 1);
    const int t  = e >> 10;        // b*D + d
    const int d  = t & (D_ - 1);
    const int b  = t >> 6;
    const int n  = (b << 10) | hw;

    const int   code = ws_idx[n];
    const float zq   = emb[code * D_ + d];
    const float zp   = z[e];
    const float diff = zq - zp;
    out[e] = zp + diff;            // zp + stopgrad(z_q - zp), same arithmetic

    sdata[threadIdx.x] = diff * diff;
    __syncthreads();
#pragma unroll
    for (int off = 128; off > 0; off >>= 1) {
        if (threadIdx.x < off) sdata[threadIdx.x] += sdata[threadIdx.x + off];
        __syncthreads();
    }
    if (threadIdx.x == 0) partial[blockIdx.x] = sdata[0];
}

// ---------------------------------------------------------------------------
// Kernel 3: deterministic final reduction -> loss = 1.25 * mean(diff^2)
// ---------------------------------------------------------------------------
__global__ __launch_bounds__(256) void vq_loss_kernel(
    const float* __restrict__ partial, int n_part, float* __restrict__ out_loss)
{
    __shared__ float sdata[256];
    float s = 0.0f;
    for (int i = threadIdx.x; i < n_part; i += 256) s += partial[i];
    sdata[threadIdx.x] = s;
    __syncthreads();
#pragma unroll
    for (int off = 128; off > 0; off >>= 1) {
        if (threadIdx.x < off) sdata[threadIdx.x] += sdata[threadIdx.x + off];
        __syncthreads();
    }
    if (threadIdx.x == 0)
        out_loss[0] = 1.25f * sdata[0] / (float)NELEM_;
}

// ---------------------------------------------------------------------------
extern "C" void kernel_launch(void* const* d_in, const int* in_sizes, int n_in,
                              void* d_out, int out_size, void* d_ws, size_t ws_size,
                              hipStream_t stream) {
    const float* z   = (const float*)d_in[0];   // [32,64,32,32] NCHW f32
    const float* emb = (const float*)d_in[1];   // [1024,64] f32
    float* out      = (float*)d_out;
    float* out_idx  = out + NELEM_;             // indices as float
    float* out_loss = out + NELEM_ + N_;        // scalar loss

    int*   ws_idx  = (int*)d_ws;                               // N ints
    float* partial = (float*)((char*)d_ws + N_ * sizeof(int)); // 8192 floats

    // 2048 waves (16 rows each), 8 waves per block -> 256 blocks
    vq_argmin_kernel<<<(N_ / 16) * 32 / 256, 256, 0, stream>>>(z, emb, out_idx, ws_idx);
    // one thread per output element
    vq_quant_kernel<<<NELEM_ / 256, 256, 0, stream>>>(z, emb, ws_idx, out, partial);
    vq_loss_kernel<<<1, 256, 0, stream>>>(partial, NELEM_ / 256, out_loss);
}